// GCNLayer_6786048328673
// MI455X (gfx1250) — compile-verified
//
#include <hip/hip_runtime.h>
#include <hip/hip_bf16.h>

// ---------------------------------------------------------------------------
// GCN layer: out = D^{-1/2} A D^{-1/2} (X W) + b   (edge-list adjacency)
//   N = 50000 nodes, E = 800000 edges, DIN = DOUT = 64, fp32 everywhere.
//
// Phase plan (all on `stream`, graph-capture safe — no mallocs/syncs):
//   1) zero degree accumulator (ws)
//   2) deg[row[e]] += 1.0           (global fp32 atomics, resolve in L2)
//   3) dinv[i] = deg>0 ? rsqrt(deg) : 0
//   4) XW = X @ W via V_WMMA_F32_16X16X4_F32  (fp32 matrix pipe)
//   5) out[i][c] = bias[c]          (bias fused into init; scatter adds on top)
//   6) per-edge wave32: out[row][c] += dinv[row]*dinv[col]*xw[col][c]
// ---------------------------------------------------------------------------

typedef __attribute__((ext_vector_type(2))) float v2f;
typedef __attribute__((ext_vector_type(8))) float v8f;

#define D_FEAT 64

// ---------------- small utility kernels ----------------

__global__ __launch_bounds__(256) void zero_f32(float* __restrict__ p, int n) {
  int i = blockIdx.x * blockDim.x + threadIdx.x;
  if (i < n) p[i] = 0.0f;
}

__global__ __launch_bounds__(256) void degree_kernel(const int* __restrict__ row,
                                                     float* __restrict__ deg, int e) {
  int i = blockIdx.x * blockDim.x + threadIdx.x;
  if (i < e) atomicAdd(&deg[row[i]], 1.0f);
}

__global__ __launch_bounds__(256) void dinv_kernel(float* __restrict__ deg_dinv, int n) {
  int i = blockIdx.x * blockDim.x + threadIdx.x;
  if (i < n) {
    float d = deg_dinv[i];
    deg_dinv[i] = (d > 0.0f) ? rsqrtf(d) : 0.0f;
  }
}

__global__ __launch_bounds__(256) void bias_init_kernel(const float* __restrict__ bias,
                                                        float* __restrict__ out, int total) {
  int i = blockIdx.x * blockDim.x + threadIdx.x;
  if (i < total) out[i] = bias[i & (D_FEAT - 1)];
}

// ---------------- WMMA fp32 GEMM: XW = X (N x 64) * W (64 x 64) ----------------
// Block = 256 threads = 8 waves. Each wave computes one 16x16 tile of XW.
// waves 0..3 -> row tile 0, col tiles 0..3 ; waves 4..7 -> row tile 1.
// Block covers 32 rows x 64 cols. N % 16 == 0, so tiles are all-or-nothing valid.
__global__ __launch_bounds__(256) void gemm_xw_wmma(const float* __restrict__ X,
                                                    const float* __restrict__ W,
                                                    float* __restrict__ XW, int n) {
  const int wave = threadIdx.x >> 5;          // 0..7
  const int lane = threadIdx.x & 31;
  const int half = lane >> 4;                 // 0: lanes 0-15, 1: lanes 16-31
  const int l    = lane & 15;

  const int mBase = (blockIdx.x * 2 + (wave >> 2)) * 16;  // 16-row tile base
  const int nBase = (wave & 3) * 16;                      // 16-col tile base
  if (mBase >= n) return;   // uniform per wave; EXEC stays all-1s inside

  // A-fragment row for this lane (16x4 f32 layout, ISA 7.12.2)
  const int m = mBase + l;

  v8f acc = {};
#pragma unroll
  for (int k = 0; k < D_FEAT; k += 4) {
    // A 16x4: lanes 0-15 carry K = k,k+1 ; lanes 16-31 carry K = k+2,k+3
    v2f a, b;
    a.x = X[m * D_FEAT + k + 2 * half + 0];
    a.y = X[m * D_FEAT + k + 2 * half + 1];
    // B 4x16: rows striped across lanes within a VGPR (transpose of A striping)
    b.x = W[(k + 2 * half + 0) * D_FEAT + nBase + l];
    b.y = W[(k + 2 * half + 1) * D_FEAT + nBase + l];
    // 8-arg form: (neg_a, A, neg_b, B, c_mod, C, reuse_a, reuse_b)
    acc = __builtin_amdgcn_wmma_f32_16x16x4_f32(false, a, false, b,
                                                (short)0, acc, false, false);
  }

  // D 16x16 f32: VGPR r -> row mBase + r + 8*half, col nBase + l
#pragma unroll
  for (int r = 0; r < 8; ++r) {
    XW[(mBase + r + 8 * half) * D_FEAT + nBase + l] = acc[r];
  }
}

// ---------------- edge scatter: one wave32 per edge ----------------
// lane handles feature columns (lane) and (lane+32): 256B coalesced gather of
// xw[col][*] and 256B coalesced GLOBAL_ATOMIC_ADD_F32 burst into out[row][*].
// Whole 45MB working set lives in the 192MB L2 -> atomics resolve at L2.
__global__ __launch_bounds__(256) void scatter_kernel(const int* __restrict__ row,
                                                      const int* __restrict__ col,
                                                      const float* __restrict__ dinv,
                                                      const float* __restrict__ xw,
                                                      float* __restrict__ out, int e) {
  const int edge = blockIdx.x * 8 + (threadIdx.x >> 5);
  if (edge >= e) return;
  const int lane = threadIdx.x & 31;

  const int r = row[edge];   // same address across wave -> broadcast load
  const int c = col[edge];
  const float w = dinv[r] * dinv[c];

  const float v0 = xw[c * D_FEAT + lane] * w;
  const float v1 = xw[c * D_FEAT + lane + 32] * w;
  atomicAdd(&out[r * D_FEAT + lane], v0);
  atomicAdd(&out[r * D_FEAT + lane + 32], v1);
}

// ---------------- launcher ----------------

extern "C" void kernel_launch(void* const* d_in, const int* in_sizes, int n_in,
                              void* d_out, int out_size, void* d_ws, size_t ws_size,
                              hipStream_t stream) {
  const float* x      = (const float*)d_in[0];
  const int*   eidx   = (const int*)d_in[1];
  const float* weight = (const float*)d_in[2];
  const float* bias   = (const float*)d_in[3];
  float*       out    = (float*)d_out;

  const int n = in_sizes[0] / D_FEAT;   // 50000
  const int e = in_sizes[1] / 2;        // 800000
  const int* row = eidx;                // edge_index[0]
  const int* col = eidx + e;            // edge_index[1]

  // workspace layout: dinv[n] (aligned to 256 elems) then xw[n*64]
  float* dinv = (float*)d_ws;
  const int dinvStride = (n + 255) & ~255;
  float* xw = dinv + dinvStride;

  // 1) zero degree accumulator
  zero_f32<<<(n + 255) / 256, 256, 0, stream>>>(dinv, n);
  // 2) degree count via fp32 atomics
  degree_kernel<<<(e + 255) / 256, 256, 0, stream>>>(row, dinv, e);
  // 3) dinv = rsqrt(deg) (in place)
  dinv_kernel<<<(n + 255) / 256, 256, 0, stream>>>(dinv, n);
  // 4) XW = X @ W  (WMMA fp32; 32 rows per block)
  gemm_xw_wmma<<<(n + 31) / 32, 256, 0, stream>>>(x, weight, xw, n);
  // 5) out = bias (broadcast init; scatter adds on top)
  bias_init_kernel<<<(n * D_FEAT + 255) / 256, 256, 0, stream>>>(bias, out, n * D_FEAT);
  // 6) edge aggregation, one wave per edge
  scatter_kernel<<<(e + 7) / 8, 256, 0, stream>>>(row, col, dinv, xw, out, e);
}